// WaveFunctionRBM_OHE_69226282877342
// MI455X (gfx1250) — compile-verified
//
#include <hip/hip_runtime.h>
#include <stdint.h>

namespace {
constexpr int   kNV   = 100000;
constexpr int   kNH   = 128;
constexpr float kXmin = -10.0f;
constexpr float kXmax =  10.0f;

constexpr int kSamplesPerIter = 4;            // rows per wave per pipeline stage
constexpr int kBlockSize      = 256;          // 8 wave32 waves per block
constexpr int kWavesPerBlock  = kBlockSize / 32;
constexpr int kGridBlocks     = 2048;
}

// Flat pointers into the LDS aperture map addr[31:0] directly onto the LDS
// address (CDNA5 ISA 10.2 aperture table), so truncation gives the byte
// address the async-to-LDS instruction wants in its VDST VGPR.
__device__ __forceinline__ uint32_t lds_addr_of(const void* p) {
    return (uint32_t)(uintptr_t)p;
}

__global__ __launch_bounds__(kBlockSize)
void rbm_ohe_kernel(const float* __restrict__ x,
                    const float* __restrict__ b,
                    const float* __restrict__ c,
                    const float* __restrict__ w,
                    float* __restrict__ out,
                    int n)
{
    // Per-wave double-buffered row stage: 2 x 4 samples x 512B per wave.
    __shared__ __align__(16) float stage[kWavesPerBlock][2][kSamplesPerIter][kNH];

    const int lane   = threadIdx.x & 31;                 // wave32
    const int wIn    = threadIdx.x >> 5;
    const int waveId = blockIdx.x * kWavesPerBlock + wIn;
    const int nWaves = gridDim.x * kWavesPerBlock;

    // Per-lane slice of c (invariant across samples) -> registers.
    const float4 cv = *reinterpret_cast<const float4*>(c + (lane << 2));

    const float inv_dx = (float)(kNV - 1) / (kXmax - kXmin);
    const int   stride = nWaves * kSamplesPerIter;

    auto binIdx = [&](int si) -> int {
        const float xv = (si < n) ? x[si] : kXmin;
        int id = (int)((xv - kXmin) * inv_dx);
        return id < 0 ? 0 : (id > kNV - 1 ? kNV - 1 : id);
    };

    // Issue one async 512B row copy per sample: 32 lanes x 16B, ASYNCcnt++.
    auto issueBatch = [&](int s, int bufSel, int* idx) {
        #pragma unroll
        for (int k = 0; k < kSamplesPerIter; ++k) {
            idx[k] = binIdx(s + k);
            const uint64_t ga =
                (uint64_t)(uintptr_t)(w + (size_t)idx[k] * kNH + (lane << 2));
            const uint32_t la =
                lds_addr_of(&stage[wIn][bufSel][k][lane << 2]);
            asm volatile("global_load_async_to_lds_b128 %0, %1, off"
                         :: "v"(la), "v"(ga)
                         : "memory");
        }
    };

    int idxCur[kSamplesPerIter];
    int idxNxt[kSamplesPerIter] = {0, 0, 0, 0};

    int s = waveId * kSamplesPerIter;
    if (s >= n) return;

    issueBatch(s, 0, idxCur);          // prime the pipeline
    int bufSel = 0;

    for (; s < n; s += stride) {
        const int  sNext   = s + stride;
        const bool hasNext = sNext < n;

        if (hasNext) {
            if (lane == 0) __builtin_prefetch(x + sNext, 0, 0);  // gfx1250 global_prefetch_b8
            issueBatch(sNext, bufSel ^ 1, idxNxt);               // 8 rows now in flight
            asm volatile("s_wait_asynccnt 0x4" ::: "memory");    // current 4 rows landed
        } else {
            asm volatile("s_wait_asynccnt 0x0" ::: "memory");
        }

        // Consume staged rows: ds_load_b128 per lane, then softplus product.
        float prod[kSamplesPerIter];
        #pragma unroll
        for (int k = 0; k < kSamplesPerIter; ++k) {
            const float4 wv = *reinterpret_cast<const float4*>(
                                  &stage[wIn][bufSel][k][lane << 2]);
            float p = 1.0f + __expf(cv.x + wv.x);
            p *=      1.0f + __expf(cv.y + wv.y);
            p *=      1.0f + __expf(cv.z + wv.z);
            p *=      1.0f + __expf(cv.w + wv.w);
            prod[k] = p;
        }

        // wave32 cross-lane product reduction: 5 xor-shuffle steps.
        #pragma unroll
        for (int m = 16; m >= 1; m >>= 1) {
            #pragma unroll
            for (int k = 0; k < kSamplesPerIter; ++k) {
                prod[k] *= __shfl_xor(prod[k], m, 32);
            }
        }

        if (lane == 0) {
            #pragma unroll
            for (int k = 0; k < kSamplesPerIter; ++k) {
                const int si = s + k;
                if (si < n) {
                    out[si] = __expf(b[idxCur[k]]) * prod[k];
                }
            }
        }

        #pragma unroll
        for (int k = 0; k < kSamplesPerIter; ++k) idxCur[k] = idxNxt[k];
        bufSel ^= 1;
    }
}

extern "C" void kernel_launch(void* const* d_in, const int* in_sizes, int n_in,
                              void* d_out, int out_size, void* d_ws, size_t ws_size,
                              hipStream_t stream)
{
    const float* x = (const float*)d_in[0];  // [N]
    const float* b = (const float*)d_in[1];  // [NV]
    const float* c = (const float*)d_in[2];  // [NH]
    const float* w = (const float*)d_in[3];  // [NV, NH]
    float* out     = (float*)d_out;          // [N]
    const int n    = in_sizes[0];

    hipLaunchKernelGGL(rbm_ohe_kernel,
                       dim3(kGridBlocks), dim3(kBlockSize), 0, stream,
                       x, b, c, w, out, n);
}